// GCNModelAE_87316685127960
// MI455X (gfx1250) — compile-verified
//
#include <hip/hip_runtime.h>
#include <math.h>

#define NNODES 10000
#define NEDGES 320000
#define DIN    256
#define DHID   256
#define DLAT   64

typedef __attribute__((ext_vector_type(2))) float v2f;
typedef __attribute__((ext_vector_type(4))) float v4f;
typedef __attribute__((ext_vector_type(8))) float v8f;

static inline int ceil_div_ll(long long a, long long b) { return (int)((a + b - 1) / b); }

// ---------------------------------------------------------------- utilities
__global__ void fill_kernel(float* __restrict__ p, float v, int n) {
  int i = blockIdx.x * blockDim.x + threadIdx.x;
  if (i < n) p[i] = v;
}

__global__ void deg_accum_kernel(const long long* __restrict__ dst, float* __restrict__ deg, int e) {
  int i = blockIdx.x * blockDim.x + threadIdx.x;
  if (i < e) atomicAdd(&deg[(int)dst[i]], 1.0f);
}

__global__ void rsqrt_kernel(float* __restrict__ p, int n) {
  int i = blockIdx.x * blockDim.x + threadIdx.x;
  if (i < n) p[i] = rsqrtf(p[i]);
}

// ---------------------------------------------------------------- GEMM (f32 WMMA)
// C[M x Nn] = A[M x K] * B[K x Nn], all row-major. One wave owns one 16x16 C tile.
// M, Nn multiples of 16; K multiple of 4. Uses V_WMMA_F32_16X16X4_F32 (exact fp32).
__global__ void gemm_wmma_f32_kernel(const float* __restrict__ A, const float* __restrict__ B,
                                     float* __restrict__ C, int M, int K, int Nn) {
  const int tilesN = Nn >> 4;
  const int tilesM = M >> 4;
  const int wave = (int)((blockIdx.x * blockDim.x + threadIdx.x) >> 5);  // wave-uniform
  if (wave >= tilesM * tilesN) return;                                   // wave-uniform exit
  const int lane  = threadIdx.x & 31;
  const int tm = wave / tilesN;
  const int tn = wave % tilesN;
  const int row   = lane & 15;          // A row / B,C column within tile
  const int khalf = (lane >> 4) << 1;   // lanes 0-15 -> K0/K1, lanes 16-31 -> K2/K3

  const float* Arow = A + (size_t)(tm * 16 + row) * K;
  const int n = tn * 16 + row;

  v8f acc = {};
  for (int kk = 0; kk < K; kk += 4) {
    v2f a, b;
    a.x = Arow[kk + khalf];
    a.y = Arow[kk + khalf + 1];
    b.x = B[(size_t)(kk + khalf)     * Nn + n];
    b.y = B[(size_t)(kk + khalf + 1) * Nn + n];
    acc = __builtin_amdgcn_wmma_f32_16x16x4_f32(false, a, false, b, (short)0, acc, false, false);
  }
  const int mbase = tm * 16 + ((lane >> 4) << 3);  // lanes 16-31 hold rows 8..15
#pragma unroll
  for (int r = 0; r < 8; ++r)
    C[(size_t)(mbase + r) * Nn + n] = acc[r];
}

// ---------------------------------------------------------------- edge scatter
// agg[dst] += h[src] * dinv[src] * dinv[dst]   (one thread per edge-feature)
__global__ void scatter_kernel(const float* __restrict__ h, const long long* __restrict__ src,
                               const long long* __restrict__ dst, const float* __restrict__ dinv,
                               float* __restrict__ agg, int D, int shift) {
  long long idx = (long long)blockIdx.x * blockDim.x + threadIdx.x;
  if (idx >= (long long)NEDGES * D) return;
  int e = (int)(idx >> shift);
  int f = (int)(idx & (D - 1));
  int s = (int)src[e];
  int d = (int)dst[e];
  float w = dinv[s] * dinv[d];
  atomicAdd(&agg[(size_t)d * D + f], h[(size_t)s * D + f] * w);
}

// ---------------------------------------------------------------- self-loop + bias + relu
__global__ void combine_kernel(const float* __restrict__ agg, const float* __restrict__ h,
                               const float* __restrict__ dinv, const float* __restrict__ bias,
                               float* __restrict__ out, int D, int shift) {
  long long idx = (long long)blockIdx.x * blockDim.x + threadIdx.x;
  if (idx >= (long long)NNODES * D) return;
  int node = (int)(idx >> shift);
  int f    = (int)(idx & (D - 1));
  float di = dinv[node];
  float v  = agg[idx] + h[idx] * di * di + bias[f];
  out[idx] = v > 0.0f ? v : 0.0f;
}

// ---------------------------------------------------------------- adj = sigmoid(S @ S^T)
// S: [NNODES x DLAT] row-major. Block = 256 threads = 8 waves.
// Block covers 128 rows (8 M-tiles, one per wave) x 64 columns.
// The shared 64-col x 64-K B-panel of S is staged in LDS once per block
// (8x cut in L2 refetch); WMMA B-fragments come from ds_load_b64.
// LDS row stride 68 floats: bank = (4n + k) mod 64 -> conflict-free halves,
// 272B rows keep 16B alignment for b128 staging stores.
#define LDSTRIDE 68
__global__ void adj_wmma_kernel(const float* __restrict__ S, float* __restrict__ out) {
  const int K = DLAT;                    // 64
  const int tilesM = NNODES / 16;        // 625

  __shared__ float Bpanel[64 * LDSTRIDE];

  const int tid    = threadIdx.x;
  const int waveIb = tid >> 5;                       // 0..7
  const int lane   = tid & 31;
  const int tm     = blockIdx.x * 8 + waveIb;        // M tile (wave-uniform)
  const int strip  = blockIdx.y;                     // 64-column strip
  const bool active = (tm < tilesM);                 // wave-uniform

  // ---- cooperative stage of B panel: rows n_local 0..63, k 0..63
  {
    const int n_local = tid >> 2;            // 4 threads per row
    const int kseg    = (tid & 3) * 16;      // 16 k's per thread
    int g = strip * 64 + n_local;
    if (g >= NNODES) g = NNODES - 1;         // clamp (stores of OOB cols are guarded)
    const float* gp = S + (size_t)g * K + kseg;
    float* lp = &Bpanel[n_local * LDSTRIDE + kseg];
#pragma unroll
    for (int i = 0; i < 4; ++i)
      *(v4f*)(lp + i * 4) = *(const v4f*)(gp + i * 4);
  }
  __syncthreads();

  if (active) {
    const int row   = lane & 15;
    const int khalf = (lane >> 4) << 1;
    const float* Arow = S + (size_t)(tm * 16 + row) * K;
    const int n0 = strip * 64 + row;

    v8f acc[4] = {};
    for (int kk = 0; kk < K; kk += 4) {
      v2f a;
      a.x = Arow[kk + khalf];
      a.y = Arow[kk + khalf + 1];
#pragma unroll
      for (int t = 0; t < 4; ++t) {
        const int n_local = row + 16 * t;
        v2f b = *(const v2f*)&Bpanel[n_local * LDSTRIDE + kk + khalf];
        acc[t] = __builtin_amdgcn_wmma_f32_16x16x4_f32(false, a, false, b, (short)0, acc[t],
                                                       false, false);
      }
    }

    const int mbase = tm * 16 + ((lane >> 4) << 3);
#pragma unroll
    for (int t = 0; t < 4; ++t) {
      int n = n0 + 16 * t;
      if (n < NNODES) {
#pragma unroll
        for (int r = 0; r < 8; ++r) {
          float v = acc[t][r];
          out[(size_t)(mbase + r) * NNODES + n] = 1.0f / (1.0f + __expf(-v));
        }
      }
    }
  }
}

// ---------------------------------------------------------------- one GCN conv layer
static void gcn_conv(const float* xin, const float* W, const float* bias,
                     int K, int Dout, int shift,
                     const long long* src, const long long* dst, const float* dinv,
                     float* tmp_h, float* tmp_agg, float* out, hipStream_t stream) {
  // h = xin @ W  (WMMA)
  {
    long long threads = (long long)(NNODES / 16) * (Dout / 16) * 32;
    gemm_wmma_f32_kernel<<<ceil_div_ll(threads, 256), 256, 0, stream>>>(
        xin, W, tmp_h, NNODES, K, Dout);
  }
  hipMemsetAsync(tmp_agg, 0, (size_t)NNODES * Dout * sizeof(float), stream);
  {
    long long total = (long long)NEDGES * Dout;
    scatter_kernel<<<ceil_div_ll(total, 256), 256, 0, stream>>>(
        tmp_h, src, dst, dinv, tmp_agg, Dout, shift);
  }
  {
    long long total = (long long)NNODES * Dout;
    combine_kernel<<<ceil_div_ll(total, 256), 256, 0, stream>>>(
        tmp_agg, tmp_h, dinv, bias, out, Dout, shift);
  }
}

// ---------------------------------------------------------------- entry point
extern "C" void kernel_launch(void* const* d_in, const int* in_sizes, int n_in,
                              void* d_out, int out_size, void* d_ws, size_t ws_size,
                              hipStream_t stream) {
  (void)in_sizes; (void)n_in; (void)out_size; (void)ws_size;

  const float*     x   = (const float*)d_in[0];
  const long long* ei  = (const long long*)d_in[1];     // int64 edge_index [2, E]
  const long long* src = ei;
  const long long* dst = ei + NEDGES;
  const float* W1 = (const float*)d_in[2];  const float* b1 = (const float*)d_in[3];
  const float* W2 = (const float*)d_in[4];  const float* b2 = (const float*)d_in[5];
  const float* W3 = (const float*)d_in[6];  const float* b3 = (const float*)d_in[7];
  const float* W4 = (const float*)d_in[8];  const float* b4 = (const float*)d_in[9];
  const float* W5 = (const float*)d_in[10]; const float* b5 = (const float*)d_in[11];

  float* out     = (float*)d_out;
  float* x_recon = out;                                       // [N, IN]
  float* adj     = out + (size_t)NNODES * DIN;                // [N, N]
  float* z       = adj + (size_t)NNODES * NNODES;             // [N, LAT]

  float* ws      = (float*)d_ws;
  float* dinv    = ws;                        // [N] (deg accum, then rsqrt in place)
  float* tmp_h   = ws + 16384;                // [N, 256]
  float* tmp_agg = tmp_h + (size_t)NNODES * 256;   // [N, 256]
  float* hbuf    = tmp_agg + (size_t)NNODES * 256; // [N, 256] reused: h1, a, s

  // deg = 1 + segment_sum(ones, dst);  dinv = rsqrt(deg)
  fill_kernel<<<ceil_div_ll(NNODES, 256), 256, 0, stream>>>(dinv, 1.0f, NNODES);
  deg_accum_kernel<<<ceil_div_ll(NEDGES, 256), 256, 0, stream>>>(dst, dinv, NEDGES);
  rsqrt_kernel<<<ceil_div_ll(NNODES, 256), 256, 0, stream>>>(dinv, NNODES);

  // encoder
  gcn_conv(x,    W1, b1, DIN,  DHID, 8, src, dst, dinv, tmp_h, tmp_agg, hbuf,    stream); // h1
  gcn_conv(hbuf, W2, b2, DHID, DLAT, 6, src, dst, dinv, tmp_h, tmp_agg, z,       stream); // z (in d_out)
  // attribute decoder
  gcn_conv(z,    W3, b3, DLAT, DHID, 8, src, dst, dinv, tmp_h, tmp_agg, hbuf,    stream); // a
  gcn_conv(hbuf, W4, b4, DHID, DIN,  8, src, dst, dinv, tmp_h, tmp_agg, x_recon, stream); // x_recon
  // structure decoder
  gcn_conv(z,    W5, b5, DLAT, DHID, 8, src, dst, dinv, tmp_h, tmp_agg, hbuf,    stream); // s

  // adj_recon = sigmoid(s @ s^T): grid = (M chunks of 128 rows) x (64-col strips)
  {
    const int tilesM = NNODES / 16;               // 625
    const int mChunks = (tilesM + 7) / 8;         // 79
    const int strips  = (tilesM + 3) / 4;         // 157
    dim3 grid(mChunks, strips);
    adj_wmma_kernel<<<grid, 256, 0, stream>>>(hbuf, adj);
  }
}